// SpatialProcessor_37263136260740
// MI455X (gfx1250) — compile-verified
//
#include <hip/hip_runtime.h>
#include <math.h>

#define Bq 2
#define Nn 1024
#define DIM 64
#define HEADS 4
#define PHD 16
#define SLOPE 0.2f

typedef __attribute__((ext_vector_type(2))) float v2f;
typedef __attribute__((ext_vector_type(8))) float v8f;
typedef int v4i __attribute__((vector_size(16)));

#define GLOBAL_AS __attribute__((address_space(1)))
#define LDS_AS    __attribute__((address_space(3)))

static __device__ __forceinline__ v8f wmma_f32(v2f a, v2f b, v8f c) {
    // V_WMMA_F32_16X16X4_F32: D = A(16x4) * B(4x16) + C(16x16), all fp32
    return __builtin_amdgcn_wmma_f32_16x16x4_f32(false, a, false, b, (short)0, c,
                                                 false, false);
}

// GLOBAL_LOAD_ASYNC_TO_LDS_B128: memory -> LDS, no VGPR data, ASYNCcnt tracked.
static __device__ __forceinline__ void async_load_b128(const float* gsrc, float* ldst) {
    __builtin_amdgcn_global_load_async_to_lds_b128(
        (GLOBAL_AS v4i*)gsrc, (LDS_AS v4i*)ldst, 0, 0);
}

// each of 256 threads issues 4 b128 async loads => 64x64 f32 chunk, 4/wave ASYNCcnt
static __device__ __forceinline__ void load_chunk_async(const float* __restrict__ src,
                                                        float* dstbuf, int t) {
    #pragma unroll
    for (int q = 0; q < 4; ++q) {
        const int lin = (t + q * 256) * 4;  // float index, 16B aligned
        async_load_b128(src + lin, dstbuf + lin);
    }
}

// ---------------------------------------------------------------- norms ----
__global__ void norm_kernel(const float* __restrict__ emb, float* __restrict__ inv) {
    int r = blockIdx.x * blockDim.x + threadIdx.x;
    if (r >= Nn) return;
    float s = 0.f;
    #pragma unroll 8
    for (int c = 0; c < DIM; ++c) { float v = emb[r * DIM + c]; s = fmaf(v, v, s); }
    inv[r] = 1.0f / fmaxf(sqrtf(s), 1e-12f);
}

// ----------------------------------------------------------------- mask ----
__global__ void mask_kernel(const float* __restrict__ emb, const float* __restrict__ inv,
                            unsigned char* __restrict__ mask) {
    __shared__ float ei[16 * DIM];
    __shared__ float ej[16 * DIM];
    const int t = threadIdx.x;
    const int i0 = blockIdx.y * 16, j0 = blockIdx.x * 16;
    #pragma unroll
    for (int q = 0; q < 4; ++q) {
        int lin = t + q * 256;
        int r = lin >> 6, c = lin & 63;
        ei[lin] = emb[(i0 + r) * DIM + c] * inv[i0 + r];
        ej[lin] = emb[(j0 + r) * DIM + c] * inv[j0 + r];
    }
    __syncthreads();
    const int i = t >> 4, j = t & 15;
    float d = 0.f;
    #pragma unroll 8
    for (int c = 0; c < DIM; ++c) d = fmaf(ei[i * DIM + c], ej[j * DIM + c], d);
    mask[(size_t)(i0 + i) * Nn + (j0 + j)] =
        ((d != 0.0f) || ((i0 + i) == (j0 + j))) ? 1 : 0;
}

// -------------------------------------------------- projection GEMM --------
// Y[2048,64] = X[2048,64] @ W[64,64]; one wave per 16-row tile, f32 WMMA.
__global__ void proj_kernel(const float* __restrict__ X, const float* __restrict__ W,
                            float* __restrict__ Y) {
    const int lane = threadIdx.x & 31;
    const int wave = threadIdx.x >> 5;
    const int r0 = (blockIdx.x * 4 + wave) * 16;
    const int m  = lane & 15;      // A: M row / B: N col / D: N col
    const int kh = lane >> 4;      // K-half selector
    v8f acc[4] = {};
    for (int k0 = 0; k0 < DIM; k0 += 4) {
        const int ka = k0 + kh * 2;
        v2f a;
        a[0] = X[(size_t)(r0 + m) * DIM + ka];
        a[1] = X[(size_t)(r0 + m) * DIM + ka + 1];
        #pragma unroll
        for (int c = 0; c < 4; ++c) {
            v2f b;
            b[0] = W[(size_t)ka * DIM + c * 16 + m];
            b[1] = W[(size_t)(ka + 1) * DIM + c * 16 + m];
            acc[c] = wmma_f32(a, b, acc[c]);
        }
    }
    #pragma unroll
    for (int c = 0; c < 4; ++c)
        #pragma unroll
        for (int r = 0; r < 8; ++r)
            Y[(size_t)(r0 + r + 8 * kh) * DIM + c * 16 + m] = acc[c][r];
}

// ------------------------------------------------ fused GATv2 attention ----
// Block: 256 threads = 8 waves, handles one (batch b, 16-row i-tile).
// xl chunks are double-buffered through LDS via async-to-LDS loads.
template <bool RELU>
__global__ void attn_kernel(const float* __restrict__ xl, const float* __restrict__ xr,
                            const float* __restrict__ att, const float* __restrict__ bias,
                            const unsigned char* __restrict__ mask,
                            float* __restrict__ out) {
    __shared__ float xl_s[2][64 * DIM];    // double-buffered [j][feat], 32KB
    __shared__ float p_s[64 * DIM];        // [j][h*16 + i]  (WMMA A layout feed)
    __shared__ float red[256];
    __shared__ float l_s[16 * HEADS];
    __shared__ float out_s[HEADS * 16 * 16];

    const int t    = threadIdx.x;
    const int b    = blockIdx.y;
    const int i0   = blockIdx.x * 16;
    const int lane = t & 31, wave = t >> 5;
    const int hh = wave & 3, jh = wave >> 2;     // WMMA role: head, j-half
    const int i  = t >> 4;                        // e-compute role: target row
    const int rem = t & 15;
    const int h  = rem >> 2, sub = rem & 3;       // head, j-subset

    const float* xlbase = xl + (size_t)(b * Nn) * DIM;

    float attv[PHD], xrv[PHD];
    const int gi = b * Nn + i0 + i;
    #pragma unroll
    for (int p = 0; p < PHD; ++p) {
        attv[p] = att[h * PHD + p];
        xrv[p]  = xr[(size_t)gi * DIM + h * PHD + p];
    }
    const unsigned char* mline = mask + (size_t)(i0 + i) * Nn;

    // ---- pass 1: masked row max ----
    float mloc = -3.0e38f;
    load_chunk_async(xlbase, xl_s[0], t);
    for (int jc = 0; jc < Nn; jc += 64) {
        const int buf = (jc >> 6) & 1;
        if (jc + 64 < Nn) {
            load_chunk_async(xlbase + (size_t)(jc + 64) * DIM, xl_s[buf ^ 1], t);
            asm volatile("s_wait_asynccnt 0x4" ::: "memory");  // current chunk done
        } else {
            asm volatile("s_wait_asynccnt 0x0" ::: "memory");
        }
        __syncthreads();
        const float* xc = xl_s[buf];
        for (int js = sub; js < 64; js += 4) {
            if (mline[jc + js]) {
                float e = 0.f;
                #pragma unroll
                for (int p = 0; p < PHD; ++p) {
                    float v = xrv[p] + xc[js * DIM + h * PHD + p];
                    v = v > 0.f ? v : SLOPE * v;
                    e = fmaf(attv[p], v, e);
                }
                mloc = fmaxf(mloc, e);
            }
        }
        __syncthreads();
    }
    red[t] = mloc;
    __syncthreads();
    const int base = t & ~3;
    const float mrow = fmaxf(fmaxf(red[base], red[base + 1]),
                             fmaxf(red[base + 2], red[base + 3]));
    __syncthreads();

    // ---- pass 2: p = exp(e-m), aggregate P @ xl with WMMA ----
    v8f acc = {};
    float lloc = 0.f;
    const int m_ = lane & 15, kh = lane >> 4;
    load_chunk_async(xlbase, xl_s[0], t);
    for (int jc = 0; jc < Nn; jc += 64) {
        const int buf = (jc >> 6) & 1;
        if (jc + 64 < Nn) {
            load_chunk_async(xlbase + (size_t)(jc + 64) * DIM, xl_s[buf ^ 1], t);
            asm volatile("s_wait_asynccnt 0x4" ::: "memory");
        } else {
            asm volatile("s_wait_asynccnt 0x0" ::: "memory");
        }
        __syncthreads();
        const float* xc = xl_s[buf];
        for (int js = sub; js < 64; js += 4) {
            float pv = 0.f;
            if (mline[jc + js]) {
                float e = 0.f;
                #pragma unroll
                for (int p = 0; p < PHD; ++p) {
                    float v = xrv[p] + xc[js * DIM + h * PHD + p];
                    v = v > 0.f ? v : SLOPE * v;
                    e = fmaf(attv[p], v, e);
                }
                pv = expf(e - mrow);
            }
            lloc += pv;
            p_s[js * DIM + h * PHD + i] = pv;
        }
        __syncthreads();
        // this wave: head hh, K-range [jh*32, jh*32+32), step 4
        #pragma unroll
        for (int kk = 0; kk < 8; ++kk) {
            const int k0 = jh * 32 + kk * 4 + kh * 2;
            v2f a, bv;
            a[0]  = p_s[k0 * DIM + hh * PHD + m_];
            a[1]  = p_s[(k0 + 1) * DIM + hh * PHD + m_];
            bv[0] = xc[k0 * DIM + hh * PHD + m_];
            bv[1] = xc[(k0 + 1) * DIM + hh * PHD + m_];
            acc = wmma_f32(a, bv, acc);
        }
        __syncthreads();
    }
    red[t] = lloc;
    __syncthreads();
    const float lsum = red[base] + red[base + 1] + red[base + 2] + red[base + 3];
    if (sub == 0) l_s[i * HEADS + h] = lsum;
    __syncthreads();

    // combine the two j-half accumulators per head, normalize, bias, write
    if (jh == 0) {
        #pragma unroll
        for (int r = 0; r < 8; ++r)
            out_s[(hh * 16 + r + 8 * kh) * 16 + m_] = acc[r];
    }
    __syncthreads();
    if (jh == 1) {
        #pragma unroll
        for (int r = 0; r < 8; ++r) {
            const int M = r + 8 * kh;
            float val = acc[r] + out_s[(hh * 16 + M) * 16 + m_];
            val = val / l_s[M * HEADS + hh] + bias[hh * PHD + m_];
            if (RELU) val = fmaxf(val, 0.f);
            out[(size_t)(b * Nn + i0 + M) * DIM + hh * PHD + m_] = val;
        }
    }
}

// -------------------------------------------------------------- launch -----
extern "C" void kernel_launch(void* const* d_in, const int* in_sizes, int n_in,
                              void* d_out, int out_size, void* d_ws, size_t ws_size,
                              hipStream_t stream) {
    (void)in_sizes; (void)n_in; (void)out_size; (void)ws_size;
    const float* x    = (const float*)d_in[0];
    const float* emb  = (const float*)d_in[1];
    const float* Wl1  = (const float*)d_in[2];
    const float* Wr1  = (const float*)d_in[3];
    const float* att1 = (const float*)d_in[4];
    const float* b1   = (const float*)d_in[5];
    const float* Wl2  = (const float*)d_in[6];
    const float* Wr2  = (const float*)d_in[7];
    const float* att2 = (const float*)d_in[8];
    const float* b2   = (const float*)d_in[9];

    char* w = (char*)d_ws;
    float*         inv  = (float*)(w);                               // 4 KB
    unsigned char* mask = (unsigned char*)(w + 4096);                // 1 MB
    float*         xlw  = (float*)(w + 4096 + 1024 * 1024);          // 512 KB
    float*         xrw  = xlw + 2048 * 64;                           // 512 KB
    float*         hbuf = xrw + 2048 * 64;                           // 512 KB

    norm_kernel<<<dim3(4), dim3(256), 0, stream>>>(emb, inv);
    mask_kernel<<<dim3(64, 64), dim3(256), 0, stream>>>(emb, inv, mask);

    // layer 1
    proj_kernel<<<dim3(32), dim3(128), 0, stream>>>(x, Wl1, xlw);
    proj_kernel<<<dim3(32), dim3(128), 0, stream>>>(x, Wr1, xrw);
    attn_kernel<true><<<dim3(Nn / 16, Bq), dim3(256), 0, stream>>>(
        xlw, xrw, att1, b1, mask, hbuf);

    // layer 2
    proj_kernel<<<dim3(32), dim3(128), 0, stream>>>(hbuf, Wl2, xlw);
    proj_kernel<<<dim3(32), dim3(128), 0, stream>>>(hbuf, Wr2, xrw);
    attn_kernel<false><<<dim3(Nn / 16, Bq), dim3(256), 0, stream>>>(
        xlw, xrw, att2, b2, mask, (float*)d_out);
}